// UpdateNetV2_70746701300263
// MI455X (gfx1250) — compile-verified
//
#include <hip/hip_runtime.h>
#include <hip/hip_bf16.h>

typedef float v2f __attribute__((ext_vector_type(2)));
typedef float v8f __attribute__((ext_vector_type(8)));

#define ROWS 8192   // b*n = 8*1024
#define NSEQ 1024
#define HDIM 256
#define DDIM 32
#define KDIM 256    // token feature dim
#define CH   128    // chunk rows in kernel 2
#define NCHUNK 8

__device__ __forceinline__ float log_sigmoid_f(float x) {
    // log(sigmoid(x)) = min(x,0) - log(1 + exp(-|x|))
    return fminf(x, 0.0f) - __logf(1.0f + __expf(-fabsf(x)));
}

__device__ __forceinline__ float silu_f(float x) {
    return x / (1.0f + __expf(-x));
}

// ---------------------------------------------------------------------------
// Kernel 1: out[8192 x N] = token[8192 x 256] @ W[256 x N], f32 WMMA 16x16x4.
// 8 waves per block; each wave computes NT adjacent 16x16 N-tiles, reusing
// one A fragment per K-step (NT x register blocking on accumulators).
// ---------------------------------------------------------------------------
template <int NT>
__global__ __launch_bounds__(256) void unv2_proj_gemm(
    const float* __restrict__ A, const float* __restrict__ W,
    float* __restrict__ out, int N) {
    const int tid  = threadIdx.x;
    const int lane = tid & 31;
    const int wave = tid >> 5;
    const int ntgroups = (N >> 4) / NT;
    const int tile = blockIdx.x * 8 + wave;
    const int m0 = (tile / ntgroups) << 4;
    const int n0 = (tile % ntgroups) * (NT * 16);

    const int Mi   = lane & 15;          // row within A tile / col within B tile
    const int ksel = (lane >> 4) * 2;    // K sub-index pair selector

    const float* arow = A + (m0 + Mi) * KDIM + ksel;       // contiguous pair
    const float* bcol = W + ksel * N + n0 + Mi;            // strided pair

    v8f c[NT];
#pragma unroll
    for (int t = 0; t < NT; ++t) c[t] = {};

    for (int k0 = 0; k0 < KDIM; k0 += 4) {
        v2f a = *(const v2f*)(arow + k0);                  // 8B aligned: even offset
#pragma unroll
        for (int t = 0; t < NT; ++t) {
            v2f b;
            b[0] = bcol[k0 * N + t * 16];
            b[1] = bcol[k0 * N + t * 16 + N];
            c[t] = __builtin_amdgcn_wmma_f32_16x16x4_f32(
                       false, a, false, b, (short)0, c[t], false, false);
        }
    }

#pragma unroll
    for (int t = 0; t < NT; ++t) {
#pragma unroll
        for (int j = 0; j < 8; ++j) {
            int row = m0 + j + (lane >> 4) * 8;
            out[row * N + n0 + t * 16 + (lane & 15)] = c[t][j];
        }
    }
}

// ---------------------------------------------------------------------------
// Kernel 2: one block per (b,h). Scan + gate in LDS, accumulate 32x32x1024
// GEMM with f32 WMMA. Output W[b,d,e,h].
// ---------------------------------------------------------------------------
__global__ __launch_bounds__(256) void unv2_head_kernel(
    const float* __restrict__ kv,    // [b*n][64]   k=[:32], v=[32:]
    const float* __restrict__ kvh,   // [b*n][512]  k_h=[:256], v_h=[256:]
    const float* __restrict__ dec,   // [b*n][32]
    const float* __restrict__ dech,  // [b*n][256]
    float* __restrict__ out) {       // [b][32][32][256]
    __shared__ float ldbuf[CH][DDIM + 1];   // local 16-row inclusive prefixes
    __shared__ float ksb[CH][DDIM + 1];
    __shared__ float vsb[CH][DDIM + 1];
    __shared__ float partial[NCHUNK][DDIM];
    __shared__ float carry[NCHUNK][DDIM];   // exclusive chunk carries
    __shared__ float subt[8][DDIM];         // 16-row subtotals within a chunk
    __shared__ float subc[8][DDIM];         // exclusive sub-carries
    __shared__ float total[DDIM];
    __shared__ float red[8][256];

    const int bh   = blockIdx.x;
    const int b    = bh >> 8;
    const int h    = bh & (HDIM - 1);
    const int tid  = threadIdx.x;
    const int lane = tid & 31;
    const int wave = tid >> 5;
    const int d    = tid & (DDIM - 1);
    const int grp  = tid >> 5;              // 8 row-groups of 16 rows

    const float* decB  = dec  + (size_t)b * NSEQ * DDIM;        // [t][d]
    const float* dechB = dech + (size_t)b * NSEQ * HDIM + h;    // [t] stride HDIM
    const float* kvB   = kv   + (size_t)b * NSEQ * 64;
    const float* kvhB  = kvh  + (size_t)b * NSEQ * 512;

    // Phase 1: per-chunk column sums of log_sigmoid(decay)
    {
        const int t0 = wave * CH;  // 8 waves x 32 lanes cover 8 chunks x 32 cols
        float s = 0.0f;
        for (int i = 0; i < CH; ++i) {
            const int t = t0 + i;
            float x = decB[t * DDIM + d] * dechB[(size_t)t * HDIM];
            s += log_sigmoid_f(x);
        }
        partial[wave][d] = s;
    }
    __syncthreads();
    if (tid < DDIM) {
        float run = 0.0f;
        for (int g = 0; g < NCHUNK; ++g) {
            carry[g][tid] = run;
            run += partial[g][tid];
        }
        total[tid] = run;
    }
    __syncthreads();

    // Per-wave output tile assignment: 4 tiles (dt,et) x 2 K-halves
    const int dt = wave >> 2;
    const int et = (wave >> 1) & 1;
    const int kh = wave & 1;
    const int Mi   = lane & 15;
    const int ksel = (lane >> 4) * 2;

    v8f c = {};

    for (int ci = 0; ci < NCHUNK; ++ci) {
        const int jc = NCHUNK - 1 - ci;  // mirrored decay chunk for suffix-cumsum
        const int i0 = grp * 16;

        // prefetch next decay chunk (emits global_prefetch_b8)
        if (jc > 0)
            __builtin_prefetch(decB + (size_t)(jc - 1) * CH * DDIM + tid, 0, 1);

        // 2a: log_sigmoid of decay chunk jc; store LOCAL inclusive prefixes
        {
            float run = 0.0f;
            for (int ii = 0; ii < 16; ++ii) {
                const int i = i0 + ii;
                const int j = jc * CH + i;
                float x = decB[j * DDIM + d] * dechB[(size_t)j * HDIM];
                run += log_sigmoid_f(x);
                ldbuf[i][d] = run;
            }
            subt[grp][d] = run;
        }
        __syncthreads();

        // 2b: 8-step register scan of the 16-row subtotals (per column)
        if (tid < DDIM) {
            float run = 0.0f;
#pragma unroll
            for (int g = 0; g < 8; ++g) {
                subc[g][tid] = run;
                run += subt[g][tid];
            }
        }
        __syncthreads();

        // 2c: build scaled K and V tiles for t-chunk ci
        for (int ii = 0; ii < 16; ++ii) {
            const int i = i0 + ii;
            const int t = ci * CH + i;
            const int lj = CH - 1 - i;   // prefix_inc at global index (n-1-t)
            float P = carry[jc][d] + subc[lj >> 4][d] + ldbuf[lj][d];
            float scale = __expf(total[d] - P);
            const float* kvrow  = kvB  + (size_t)t * 64;
            const float* kvhrow = kvhB + (size_t)t * 512 + h;
            float xk = kvrow[d]        * kvhrow[0];
            float xv = kvrow[DDIM + d] * kvhrow[HDIM];
            ksb[i][d] = silu_f(xk) * scale;
            vsb[i][d] = silu_f(xv);
        }
        __syncthreads();

        // 2d: WMMA accumulate: C[16x16] += Ksub^T * Vsub over this chunk
        {
            const int rbase = kh * 64;
            for (int s = 0; s < 64; s += 4) {
                const int r = rbase + s + ksel;
                v2f a, bb;
                a[0]  = ksb[r][dt * 16 + Mi];
                a[1]  = ksb[r + 1][dt * 16 + Mi];
                bb[0] = vsb[r][et * 16 + Mi];
                bb[1] = vsb[r + 1][et * 16 + Mi];
                c = __builtin_amdgcn_wmma_f32_16x16x4_f32(
                        false, a, false, bb, (short)0, c, false, false);
            }
        }
        __syncthreads();
    }

    // Phase 3: reduce the two K-halves and store bdeh output
#pragma unroll
    for (int j = 0; j < 8; ++j) red[wave][j * 32 + lane] = c[j];
    __syncthreads();
    if (kh == 0) {
#pragma unroll
        for (int j = 0; j < 8; ++j) {
            float val = c[j] + red[wave + 1][j * 32 + lane];
            const int dd = dt * 16 + j + (lane >> 4) * 8;
            const int ee = et * 16 + (lane & 15);
            out[(((size_t)b * DDIM + dd) * DDIM + ee) * HDIM + h] = val;
        }
    }
}

extern "C" void kernel_launch(void* const* d_in, const int* in_sizes, int n_in,
                              void* d_out, int out_size, void* d_ws, size_t ws_size,
                              hipStream_t stream) {
    const float* token    = (const float*)d_in[0];
    const float* W_kv     = (const float*)d_in[1];
    const float* W_kvh    = (const float*)d_in[2];
    const float* W_decay  = (const float*)d_in[3];
    const float* W_decayh = (const float*)d_in[4];
    float* out = (float*)d_out;

    float* kv_o   = (float*)d_ws;                 // 8192 x 64
    float* kvh_o  = kv_o  + (size_t)ROWS * 64;    // 8192 x 512
    float* dec_o  = kvh_o + (size_t)ROWS * 512;   // 8192 x 32
    float* dech_o = dec_o + (size_t)ROWS * 32;    // 8192 x 256

    // blocks = (ROWS/16) * (N/16/NT) / 8 waves
    unv2_proj_gemm<4><<<(ROWS / 16) * (64  / 16 / 4) / 8, 256, 0, stream>>>(token, W_kv,     kv_o,   64);
    unv2_proj_gemm<4><<<(ROWS / 16) * (512 / 16 / 4) / 8, 256, 0, stream>>>(token, W_kvh,    kvh_o,  512);
    unv2_proj_gemm<2><<<(ROWS / 16) * (32  / 16 / 2) / 8, 256, 0, stream>>>(token, W_decay,  dec_o,  32);
    unv2_proj_gemm<4><<<(ROWS / 16) * (256 / 16 / 4) / 8, 256, 0, stream>>>(token, W_decayh, dech_o, 256);

    unv2_head_kernel<<<8 * HDIM, 256, 0, stream>>>(kv_o, kvh_o, dec_o, dech_o, out);
}